// RGCN_87514253623555
// MI455X (gfx1250) — compile-verified
//
#include <hip/hip_runtime.h>
#include <hip/hip_bf16.h>

// RGCN (3-layer, basis-decomposed, norm='both') for MI455X / gfx1250.
// Full-fp32 math via V_WMMA_F32_16X16X4_F32 (wave32 WMMA).
//
// Fixed problem shape from reference: N=50000, D_IN=D_H=128, D_OUT=64,
// R=4, E=400000, NB=2.  N%16==0 and all dims %64==0 are relied upon.
//
// Workspace layout (floats), total ~79.1 MB:
//   W0[4*128*128] W1[4*128*128] W2[4*128*64] normS[4*N] normD[4*N]
//   hA[N*128] hB[N*128] hr[N*128]

typedef float v2f __attribute__((ext_vector_type(2)));
typedef float v8f __attribute__((ext_vector_type(8)));

// ---------------------------------------------------------------- utilities
__global__ void rgcn_zero_f32(float* __restrict__ p, int n) {
  int i = blockIdx.x * blockDim.x + threadIdx.x;
  if (i < n) p[i] = 0.0f;
}

__global__ void rgcn_relu_f32(float* __restrict__ p, int n) {
  int i = blockIdx.x * blockDim.x + threadIdx.x;
  if (i < n) p[i] = fmaxf(p[i], 0.0f);
}

// W[r,i,o] = sum_b coeff[r,b] * basis[b,i,o]
__global__ void rgcn_build_w(const float* __restrict__ basis,
                             const float* __restrict__ coeff,
                             float* __restrict__ W,
                             int nb, int di, int dout) {
  const int r   = blockIdx.y;
  const int idx = blockIdx.x * blockDim.x + threadIdx.x;  // over di*dout
  const int sz  = di * dout;
  if (idx >= sz) return;
  float s = 0.0f;
  for (int b = 0; b < nb; ++b)
    s += coeff[r * nb + b] * basis[(size_t)b * sz + idx];
  W[(size_t)r * sz + idx] = s;
}

// degree histograms (float counts are exact up to 2^24)
__global__ void rgcn_degrees(const int* __restrict__ src,
                             const int* __restrict__ dst,
                             float* __restrict__ degS,
                             float* __restrict__ degD,
                             int E, int N) {
  const int r = blockIdx.y;
  const int e = blockIdx.x * blockDim.x + threadIdx.x;
  if (e >= E) return;
  atomicAdd(&degS[(size_t)r * N + src[(size_t)r * E + e]], 1.0f);
  atomicAdd(&degD[(size_t)r * N + dst[(size_t)r * E + e]], 1.0f);
}

__global__ void rgcn_make_norm(float* __restrict__ deg, int n) {
  int i = blockIdx.x * blockDim.x + threadIdx.x;
  if (i < n) deg[i] = rsqrtf(fmaxf(deg[i], 1.0f));
}

// ---------------------------------------------------------------- WMMA GEMM
// hr[n, :] = (h[n, :] * nsrc[n]) @ W            (W is K x DOtot, row-major)
//
// Block = 128 threads = 4 wave32s. Each wave computes a 16-row x 64-col strip
// using 4 accumulator tiles of V_WMMA_F32_16X16X4_F32. gridDim.y selects a
// 64-column group; its W slab is staged in LDS (pitch 80 floats so the two
// half-waves, reading rows k and k+2, use disjoint LDS bank groups).
#define BPITCH 80
__global__ void __launch_bounds__(128)
rgcn_gemm(const float* __restrict__ h,     // [N, K]
          const float* __restrict__ W,     // [K, DOtot]
          const float* __restrict__ nsrc,  // [N]
          float* __restrict__ hr,          // [N, DOtot]
          int N, int K, int DOtot) {
  __shared__ float Bs[128 * BPITCH];       // K<=128, 64 cols staged, 40 KB

  const int tid     = threadIdx.x;
  const int wave    = tid >> 5;
  const int lane    = tid & 31;
  const int colBase = blockIdx.y * 64;

  // cooperative staged load of W[:, colBase:colBase+64] (float4, coalesced)
  for (int idx = tid; idx < K * 16; idx += 128) {
    const int k  = idx >> 4;
    const int c4 = (idx & 15) << 2;
    const float4 v = *(const float4*)&W[(size_t)k * DOtot + colBase + c4];
    float* d = &Bs[k * BPITCH + c4];
    d[0] = v.x; d[1] = v.y; d[2] = v.z; d[3] = v.w;
  }
  __syncthreads();

  const int row0 = (blockIdx.x * 4 + wave) * 16;    // wave-uniform guard
  if (row0 >= N) return;                            // (EXEC all-ones inside)

  const int m   = lane & 15;                        // M index / N index
  const int hi  = lane >> 4;                        // half-wave: K pair select
  const int row = row0 + m;
  const float ns = nsrc[row];
  const float* arow = h + (size_t)row * K + 2 * hi; // this lane's K slots

  v8f acc0 = {}, acc1 = {}, acc2 = {}, acc3 = {};

  for (int k0 = 0; k0 < K; k0 += 4) {
    // A fragment: lane holds A[row][k0+2*hi], A[row][k0+2*hi+1]
    const float2 av = *(const float2*)(arow + k0);
    v2f a; a.x = av.x * ns; a.y = av.y * ns;

    // B fragments: lane holds W[k0+2*hi][col], W[k0+2*hi+1][col]
    const float* b0p = &Bs[(k0 + 2 * hi) * BPITCH + m];
    v2f b0; b0.x = b0p[ 0]; b0.y = b0p[BPITCH +  0];
    v2f b1; b1.x = b0p[16]; b1.y = b0p[BPITCH + 16];
    v2f b2; b2.x = b0p[32]; b2.y = b0p[BPITCH + 32];
    v2f b3; b3.x = b0p[48]; b3.y = b0p[BPITCH + 48];

    acc0 = __builtin_amdgcn_wmma_f32_16x16x4_f32(false, a, false, b0,
                                                 (short)0, acc0, false, false);
    acc1 = __builtin_amdgcn_wmma_f32_16x16x4_f32(false, a, false, b1,
                                                 (short)0, acc1, false, false);
    acc2 = __builtin_amdgcn_wmma_f32_16x16x4_f32(false, a, false, b2,
                                                 (short)0, acc2, false, false);
    acc3 = __builtin_amdgcn_wmma_f32_16x16x4_f32(false, a, false, b3,
                                                 (short)0, acc3, false, false);
  }

  // C/D layout: VGPR v -> M = v + 8*hi (lanes 16-31 hold rows 8..15), N = m.
  #pragma unroll
  for (int v = 0; v < 8; ++v) {
    float* orow = hr + (size_t)(row0 + v + 8 * hi) * DOtot + colBase + m;
    orow[ 0] = acc0[v];
    orow[16] = acc1[v];
    orow[32] = acc2[v];
    orow[48] = acc3[v];
  }
}

// ------------------------------------------------------------- edge scatter
// out[dst[e], :] += hr[src[e], :] * norm_dst[dst[e]]   (one wave per edge)
__global__ void rgcn_scatter(const float* __restrict__ hr,
                             const int* __restrict__ src,
                             const int* __restrict__ dst,
                             const float* __restrict__ ndst,
                             float* __restrict__ out,
                             int E, int DO) {
  const int wid  = (blockIdx.x * blockDim.x + threadIdx.x) >> 5;
  const int lane = threadIdx.x & 31;
  if (wid >= E) return;
  const int s = src[wid];
  const int d = dst[wid];
  const float nd = ndst[d];
  const float* hrow = hr + (size_t)s * DO;
  float* orow = out + (size_t)d * DO;
  for (int j = lane; j < DO; j += 32)        // coalesced 128 B per iteration
    atomicAdd(&orow[j], hrow[j] * nd);
}

// ------------------------------------------------------------------- driver
extern "C" void kernel_launch(void* const* d_in, const int* in_sizes, int n_in,
                              void* d_out, int out_size, void* d_ws, size_t ws_size,
                              hipStream_t stream) {
  const float* x      = (const float*)d_in[0];
  const int*   src    = (const int*)  d_in[1];
  const int*   dst    = (const int*)  d_in[2];
  const float* basis0 = (const float*)d_in[3];
  const float* coeff0 = (const float*)d_in[4];
  const float* basis1 = (const float*)d_in[5];
  const float* coeff1 = (const float*)d_in[6];
  const float* basis2 = (const float*)d_in[7];
  const float* coeff2 = (const float*)d_in[8];

  const int DIN = 128, DH = 128, DOUT = 64, R = 4, NB = 2;
  const int N = in_sizes[0] / DIN;   // 50000
  const int E = in_sizes[1] / R;     // 400000

  float* ws = (float*)d_ws;
  float* W0 = ws;
  float* W1 = W0 + (size_t)R * DIN * DH;
  float* W2 = W1 + (size_t)R * DH * DH;
  float* nS = W2 + (size_t)R * DH * DOUT;
  float* nD = nS + (size_t)R * N;
  float* hA = nD + (size_t)R * N;
  float* hB = hA + (size_t)N * DH;
  float* hr = hB + (size_t)N * DH;

  const int T = 256;
  auto cdiv = [](int a, int b) { return (a + b - 1) / b; };

  // --- graph norms (identical for all layers) ---
  rgcn_zero_f32<<<cdiv(R * N, T), T, 0, stream>>>(nS, R * N);
  rgcn_zero_f32<<<cdiv(R * N, T), T, 0, stream>>>(nD, R * N);
  rgcn_degrees<<<dim3(cdiv(E, T), R), T, 0, stream>>>(src, dst, nS, nD, E, N);
  rgcn_make_norm<<<cdiv(R * N, T), T, 0, stream>>>(nS, R * N);
  rgcn_make_norm<<<cdiv(R * N, T), T, 0, stream>>>(nD, R * N);

  // --- basis-combined weights ---
  rgcn_build_w<<<dim3(cdiv(DIN * DH, T), R), T, 0, stream>>>(basis0, coeff0, W0, NB, DIN, DH);
  rgcn_build_w<<<dim3(cdiv(DH * DH, T),  R), T, 0, stream>>>(basis1, coeff1, W1, NB, DH, DH);
  rgcn_build_w<<<dim3(cdiv(DH * DOUT, T), R), T, 0, stream>>>(basis2, coeff2, W2, NB, DH, DOUT);

  const int rowBlocks = cdiv(cdiv(N, 16), 4);   // 4 row-tiles per block

  // --- layer 0: x -> hA (d_out=128), ReLU ---
  rgcn_zero_f32<<<cdiv(N * DH, T), T, 0, stream>>>(hA, N * DH);
  for (int r = 0; r < R; ++r) {
    rgcn_gemm<<<dim3(rowBlocks, DH / 64), 128, 0, stream>>>(
        x, W0 + (size_t)r * DIN * DH, nS + (size_t)r * N, hr, N, DIN, DH);
    rgcn_scatter<<<cdiv(E * 32, T), T, 0, stream>>>(
        hr, src + (size_t)r * E, dst + (size_t)r * E, nD + (size_t)r * N, hA, E, DH);
  }
  rgcn_relu_f32<<<cdiv(N * DH, T), T, 0, stream>>>(hA, N * DH);

  // --- layer 1: hA -> hB (d_out=128), ReLU ---
  rgcn_zero_f32<<<cdiv(N * DH, T), T, 0, stream>>>(hB, N * DH);
  for (int r = 0; r < R; ++r) {
    rgcn_gemm<<<dim3(rowBlocks, DH / 64), 128, 0, stream>>>(
        hA, W1 + (size_t)r * DH * DH, nS + (size_t)r * N, hr, N, DH, DH);
    rgcn_scatter<<<cdiv(E * 32, T), T, 0, stream>>>(
        hr, src + (size_t)r * E, dst + (size_t)r * E, nD + (size_t)r * N, hB, E, DH);
  }
  rgcn_relu_f32<<<cdiv(N * DH, T), T, 0, stream>>>(hB, N * DH);

  // --- layer 2: hB -> d_out (d_out=64), no activation ---
  float* out = (float*)d_out;
  rgcn_zero_f32<<<cdiv(out_size, T), T, 0, stream>>>(out, out_size);
  for (int r = 0; r < R; ++r) {
    rgcn_gemm<<<dim3(rowBlocks, DOUT / 64), 128, 0, stream>>>(
        hB, W2 + (size_t)r * DH * DOUT, nS + (size_t)r * N, hr, N, DH, DOUT);
    rgcn_scatter<<<cdiv(E * 32, T), T, 0, stream>>>(
        hr, src + (size_t)r * E, dst + (size_t)r * E, nD + (size_t)r * N, out, E, DOUT);
  }
}